// Model_47081431499056
// MI455X (gfx1250) — compile-verified
//
#include <hip/hip_runtime.h>
#include <hip/hip_bf16.h>
#include <limits.h>

// ---------------------------------------------------------------------------
// GCN (2 layers) + global max pool + MLP for MI455X (gfx1250, wave32).
// Memory-bound: edge gather/scatter dominates (~1.7 GB through L2).
// The 100000x64x64 GEMM uses V_WMMA_F32_16X16X4_F32 (full f32 precision).
// ---------------------------------------------------------------------------

#define N_NODES   100000
#define N_EDGES   1600000
#define NUM_G     128
#define C         64          // feature width
#define NC        (N_NODES * C)

typedef __attribute__((ext_vector_type(2))) float v2f;
typedef __attribute__((ext_vector_type(8))) float v8f;

// ---- order-preserving float<->int encoding for atomicMax-based segment_max
__device__ __forceinline__ int  fenc(float f) {
    int i = __float_as_int(f);
    return (i >= 0) ? i : (i ^ 0x7FFFFFFF);
}
__device__ __forceinline__ float fdec(int i) {
    return __int_as_float((i >= 0) ? i : (i ^ 0x7FFFFFFF));
}

// ---------------------------------------------------------------------------
__global__ void k_fill_f32(float* __restrict__ p, int n, float v) {
    int i = blockIdx.x * blockDim.x + threadIdx.x;
    if (i < n) p[i] = v;
}
__global__ void k_fill_i32(int* __restrict__ p, int n, int v) {
    int i = blockIdx.x * blockDim.x + threadIdx.x;
    if (i < n) p[i] = v;
}

// deg[dst] += edge_attr  (segment_sum over edges)
__global__ void k_deg(const int* __restrict__ dst, const float* __restrict__ ea,
                      float* __restrict__ deg) {
    int e = blockIdx.x * blockDim.x + threadIdx.x;
    if (e < N_EDGES) atomicAdd(deg + dst[e], ea[e]);
}

// dis = rsqrt(deg+1) (in-place over deg buffer), selfc = dis*dis
__global__ void k_norm(float* __restrict__ deg_dis, float* __restrict__ selfc) {
    int i = blockIdx.x * blockDim.x + threadIdx.x;
    if (i >= N_NODES) return;
    float d  = deg_dis[i] + 1.0f;
    float di = rsqrtf(d);
    deg_dis[i] = di;
    selfc[i]   = di * di;
}

// h1 = pos @ W1   (N x 3) @ (3 x 64); one thread per output element
__global__ void k_h1(const float* __restrict__ pos, const float* __restrict__ W1,
                     float* __restrict__ h1) {
    int i = blockIdx.x * blockDim.x + threadIdx.x;
    if (i >= NC) return;
    int n = i >> 6, c = i & 63;
    const float* p = pos + n * 3;
    h1[i] = p[0] * W1[c] + p[1] * W1[64 + c] + p[2] * W1[128 + c];
}

// agg[dst] += (dis[src]*ea*dis[dst]) * H[src]  — 16 lanes/edge, float4 each
__global__ void k_scatter(const float* __restrict__ H,
                          const int* __restrict__ src, const int* __restrict__ dst,
                          const float* __restrict__ ea, const float* __restrict__ dis,
                          float* __restrict__ agg) {
    long gid = (long)blockIdx.x * blockDim.x + threadIdx.x;
    long e   = gid >> 4;
    if (e >= N_EDGES) return;
    int q = (int)(gid & 15);
    int s = src[e], d = dst[e];
    float coef = dis[s] * ea[e] * dis[d];
    const float4 hv = *(const float4*)(H + (long)s * C + q * 4);
    float* ap = agg + (long)d * C + q * 4;
    atomicAdd(ap + 0, coef * hv.x);
    atomicAdd(ap + 1, coef * hv.y);
    atomicAdd(ap + 2, coef * hv.z);
    atomicAdd(ap + 3, coef * hv.w);
}

// x = relu(agg + h*selfc + b) written in-place over h; agg re-zeroed for reuse
__global__ void k_combine_relu(float* __restrict__ h, float* __restrict__ agg,
                               const float* __restrict__ selfc,
                               const float* __restrict__ b) {
    int i = blockIdx.x * blockDim.x + threadIdx.x;
    if (i >= NC) return;
    int n = i >> 6, c = i & 63;
    float v = agg[i] + h[i] * selfc[n] + b[c];
    h[i]   = fmaxf(v, 0.0f);
    agg[i] = 0.0f;
}

// Y[100000x64] = X[100000x64] @ W[64x64] via V_WMMA_F32_16X16X4_F32.
// Block = 128 threads = 4 waves; blockIdx = M-tile (16 rows), wave = N-tile.
// A layout: lane half h, VGPR v holds K = v + 2h (M = lane&15).
// B layout: VGPR v, half h holds K = v + 2h (N = lane&15).
// C layout: VGPR r holds row r (lanes 0-15) and row r+8 (lanes 16-31).
__global__ void k_gemm_wmma(const float* __restrict__ X, const float* __restrict__ W,
                            float* __restrict__ Y) {
    int wave = threadIdx.x >> 5;       // 0..3 -> N tile
    int lane = threadIdx.x & 31;
    int half = lane >> 4;              // 0 or 1
    int lm   = lane & 15;
    long row = (long)blockIdx.x * 16 + lm;      // A row for this lane
    int  col = wave * 16 + lm;                  // B/C column for this lane

    v8f c = {};
    const float* xr = X + row * C;
#pragma unroll
    for (int k0 = 0; k0 < C; k0 += 4) {
        v2f a = *(const v2f*)(xr + k0 + 2 * half);     // K=k0+2h, k0+2h+1 (8B aligned)
        v2f b;
        b.x = W[(k0 + 2 * half) * C + col];
        b.y = W[(k0 + 2 * half + 1) * C + col];
        c = __builtin_amdgcn_wmma_f32_16x16x4_f32(
                /*neg_a=*/false, a, /*neg_b=*/false, b,
                /*c_mod=*/(short)0, c, /*reuse_a=*/false, /*reuse_b=*/false);
    }
#pragma unroll
    for (int r = 0; r < 8; ++r) {
        long orow = (long)blockIdx.x * 16 + r + 8 * half;
        Y[orow * C + col] = c[r];
    }
}

// v = agg + h2*selfc + b2 ; pool[batch[n]] = max(pool, v) via int-encoded atomicMax
__global__ void k_pool_max(const float* __restrict__ h2, const float* __restrict__ agg,
                           const float* __restrict__ selfc, const float* __restrict__ b2,
                           const int* __restrict__ batch, int* __restrict__ pool) {
    int i = blockIdx.x * blockDim.x + threadIdx.x;
    if (i >= NC) return;
    int n = i >> 6, c = i & 63;
    float v = agg[i] + h2[i] * selfc[n] + b2[c];
    atomicMax(pool + batch[n] * C + c, fenc(v));
}

// Per-graph MLP [64]->[64]->[64]->[10]; 128 blocks x 64 threads, LDS ping-pong
__global__ void k_mlp(const int* __restrict__ pool,
                      const float* __restrict__ Wm1, const float* __restrict__ bm1,
                      const float* __restrict__ Wm2, const float* __restrict__ bm2,
                      const float* __restrict__ Wm3, const float* __restrict__ bm3,
                      float* __restrict__ out) {
    __shared__ float s0[64];
    __shared__ float s1[64];
    int b = blockIdx.x, t = threadIdx.x;
    s0[t] = fdec(pool[b * C + t]);
    __syncthreads();
    float acc = bm1[t];
#pragma unroll 8
    for (int k = 0; k < 64; ++k) acc += s0[k] * Wm1[k * 64 + t];
    s1[t] = fmaxf(acc, 0.0f);
    __syncthreads();
    acc = bm2[t];
#pragma unroll 8
    for (int k = 0; k < 64; ++k) acc += s1[k] * Wm2[k * 64 + t];
    s0[t] = fmaxf(acc, 0.0f);   // safe: all s0 reads finished before prior barrier
    __syncthreads();
    if (t < 10) {
        float o = bm3[t];
#pragma unroll 8
        for (int k = 0; k < 64; ++k) o += s0[k] * Wm3[k * 10 + t];
        out[b * 10 + t] = o;
    }
}

// ---------------------------------------------------------------------------
extern "C" void kernel_launch(void* const* d_in, const int* in_sizes, int n_in,
                              void* d_out, int out_size, void* d_ws, size_t ws_size,
                              hipStream_t stream) {
    const float* pos   = (const float*)d_in[0];
    const int*   eidx  = (const int*)  d_in[1];   // [2, E] flat
    const float* ea    = (const float*)d_in[2];
    const int*   batch = (const int*)  d_in[3];
    const float* W1    = (const float*)d_in[4];
    const float* b1    = (const float*)d_in[5];
    const float* W2    = (const float*)d_in[6];
    const float* b2    = (const float*)d_in[7];
    const float* Wm1   = (const float*)d_in[8];
    const float* bm1   = (const float*)d_in[9];
    const float* Wm2   = (const float*)d_in[10];
    const float* bm2   = (const float*)d_in[11];
    const float* Wm3   = (const float*)d_in[12];
    const float* bm3   = (const float*)d_in[13];
    float* out = (float*)d_out;

    const int* src = eidx;
    const int* dst = eidx + N_EDGES;

    // workspace layout (f32 unless noted): dis | selfc | h1/x1 | agg | h2 | pool(i32)
    float* dis   = (float*)d_ws;              // deg sum, then dis (in place)
    float* selfc = dis   + N_NODES;
    float* h1    = selfc + N_NODES;           // becomes x1 in place
    float* agg   = h1    + NC;                // reused for both layers
    float* h2    = agg   + NC;
    int*   pool  = (int*)(h2 + NC);           // 128*64 ints

    const int T = 256;
    // init
    k_fill_f32<<<(N_NODES + T - 1) / T, T, 0, stream>>>(dis, N_NODES, 0.0f);
    k_fill_f32<<<(NC + T - 1) / T, T, 0, stream>>>(agg, NC, 0.0f);
    k_fill_i32<<<(NUM_G * C + T - 1) / T, T, 0, stream>>>(pool, NUM_G * C, INT_MIN);

    // degree + normalization
    k_deg<<<(N_EDGES + T - 1) / T, T, 0, stream>>>(dst, ea, dis);
    k_norm<<<(N_NODES + T - 1) / T, T, 0, stream>>>(dis, selfc);

    // layer 1
    k_h1<<<(NC + T - 1) / T, T, 0, stream>>>(pos, W1, h1);
    {
        long thr = (long)N_EDGES * 16;
        k_scatter<<<(unsigned)((thr + T - 1) / T), T, 0, stream>>>(h1, src, dst, ea, dis, agg);
    }
    k_combine_relu<<<(NC + T - 1) / T, T, 0, stream>>>(h1, agg, selfc, b1); // x1 in h1, agg->0

    // layer 2: h2 = x1 @ W2 via WMMA
    k_gemm_wmma<<<N_NODES / 16, 128, 0, stream>>>(h1, W2, h2);
    {
        long thr = (long)N_EDGES * 16;
        k_scatter<<<(unsigned)((thr + T - 1) / T), T, 0, stream>>>(h2, src, dst, ea, dis, agg);
    }

    // combine + global max pool
    k_pool_max<<<(NC + T - 1) / T, T, 0, stream>>>(h2, agg, selfc, b2, batch, pool);

    // per-graph MLP
    k_mlp<<<NUM_G, 64, 0, stream>>>(pool, Wm1, bm1, Wm2, bm2, Wm3, bm3, out);
}